// MeanFieldMultiDimensionalLogisticRegression_21646635172595
// MI455X (gfx1250) — compile-verified
//
#include <hip/hip_runtime.h>
#include <math.h>
#include <stdint.h>

// Problem constants (match reference)
#define N_PTS 16384
#define DIM   1024
#define NS    256

typedef float v2f __attribute__((ext_vector_type(2)));
typedef float v8f __attribute__((ext_vector_type(8)));

#define LDS_STRIDE       36                       // 32 floats + pad: 16B-aligned rows, conflict-free banks
#define CHUNK_FLOATS     (16 * LDS_STRIDE)        // one 16x32 staged X chunk (576 floats)
#define NBUF             4                        // async pipeline depth (prefetch 3 ahead)
#define WAVE_LDS         (NBUF * CHUNK_FLOATS + 32)  // 4 chunk buffers + 16 mean + 16 var
#define WAVES_PER_BLOCK  8
#define NCHUNK           (DIM / 32)

__device__ __forceinline__ float sigmoidf_fast(float x) {
    return 1.0f / (1.0f + __expf(-x));
}
__device__ __forceinline__ float4 affine4(float sd, float4 zz, float mu) {
    float4 r;
    r.x = fmaf(sd, zz.x, mu);
    r.y = fmaf(sd, zz.y, mu);
    r.z = fmaf(sd, zz.z, mu);
    r.w = fmaf(sd, zz.w, mu);
    return r;
}
__device__ __forceinline__ float4 sig4(float4 a) {
    float4 r;
    r.x = sigmoidf_fast(a.x);
    r.y = sigmoidf_fast(a.y);
    r.z = sigmoidf_fast(a.z);
    r.w = sigmoidf_fast(a.w);
    return r;
}

// Stage one 16x32 f32 chunk of X into an LDS buffer with 4 async b128 transfers.
// GV mode: per-lane 64-bit global address, per-lane 32-bit LDS byte address
// (low 32 bits of the generic LDS pointer == LDS offset, per aperture rules).
__device__ __forceinline__ void stage_chunk_async(const float* __restrict__ X,
                                                  int tileRow, int k0, int lane,
                                                  const float* buf) {
    #pragma unroll
    for (int i = 0; i < 4; ++i) {
        const int f  = i * 128 + lane * 4;    // flat index in 16x32 row-major chunk
        const int r  = f >> 5;
        const int kk = f & 31;
        const uint64_t g = (uint64_t)(uintptr_t)(X + (size_t)(tileRow + r) * DIM + (k0 + kk));
        const uint32_t l = (uint32_t)(uintptr_t)(buf + r * LDS_STRIDE + kk);
        asm volatile("global_load_async_to_lds_b128 %0, %1, off"
                     :: "v"(l), "v"(g) : "memory");
    }
}

// Tiny pre-pass: Ew[d] = exp(w_log_var[d]) into workspace (4 KB, L2-resident).
__global__ void expw_kernel(const float* __restrict__ wlv, float* __restrict__ Ew) {
    int i = blockIdx.x * blockDim.x + threadIdx.x;
    if (i < DIM) Ew[i] = __expf(wlv[i]);
}

// Fused main pass: one wave32 per 16-row tile of X.
//   X staged via async global->LDS DMA (ASYNCcnt pipeline, depth 3),
//   mean/var reductions via V_WMMA_F32_16X16X4_F32 (exact f32 matrix path),
//   epilogue computes act = std*z + mean and Y = sigmoid(act).
__global__ __launch_bounds__(256)
void mf_logreg_kernel(const float* __restrict__ X, const float* __restrict__ z,
                      const float* __restrict__ wmu, const float* __restrict__ Ew,
                      float* __restrict__ Y, float* __restrict__ Act) {
    __shared__ float lds[WAVES_PER_BLOCK * WAVE_LDS];

    const int tid     = threadIdx.x;
    const int wave    = tid >> 5;
    const int lane    = tid & 31;
    const int rowA    = lane & 15;   // M index this lane feeds to WMMA A
    const int hi      = lane >> 4;   // 0: K pair {0,1}; 1: K pair {2,3}
    const int tileRow = blockIdx.x * (WAVES_PER_BLOCK * 16) + wave * 16;

    float* bufs  = &lds[wave * WAVE_LDS];           // NBUF chunk buffers
    float* meanv = bufs + NBUF * CHUNK_FLOATS;
    float* varv  = meanv + 16;

    v8f accMu  = {};
    v8f accVar = {};

    // Prime the async pipeline: chunks 0..2 -> buffers 0..2.
    stage_chunk_async(X, tileRow, 0 * 32, lane, bufs + 0 * CHUNK_FLOATS);
    stage_chunk_async(X, tileRow, 1 * 32, lane, bufs + 1 * CHUNK_FLOATS);
    stage_chunk_async(X, tileRow, 2 * 32, lane, bufs + 2 * CHUNK_FLOATS);

    for (int c = 0; c < NCHUNK; ++c) {
        const int k0 = c * 32;

        // Prefetch chunk c+3 (wrap-around at the tail: harmless cached re-read,
        // keeps the wait immediate uniform).
        int pfc = c + 3;
        if (pfc >= NCHUNK) pfc -= NCHUNK;
        stage_chunk_async(X, tileRow, pfc * 32, lane,
                          bufs + ((c + 3) & (NBUF - 1)) * CHUNK_FLOATS);

        // Async loads complete in order: outstanding <= 12 (3 chunks in flight)
        // guarantees chunk c's 4 transfers have landed in LDS.
        asm volatile("s_wait_asynccnt 12" ::: "memory");

        const float* cur = bufs + (c & (NBUF - 1)) * CHUNK_FLOATS;

        // 8 x (K=4) f32 WMMA micro-steps per 32-k chunk, two accumulator chains.
        #pragma unroll
        for (int j = 0; j < 8; ++j) {
            const int kb = 4 * j + 2 * hi;
            v2f a  = *(const v2f*)(cur + rowA * LDS_STRIDE + kb);   // A: X tile
            v2f bm = *(const v2f*)(wmu + k0 + kb);                  // B: w_mu broadcast over N
            v2f be = *(const v2f*)(Ew  + k0 + kb);                  // B: exp(w_log_var) broadcast
            v2f a2 = a * a;                                         // X^2 for the variance chain
            accMu  = __builtin_amdgcn_wmma_f32_16x16x4_f32(
                         false, a,  false, bm, (short)0, accMu,  false, false);
            accVar = __builtin_amdgcn_wmma_f32_16x16x4_f32(
                         false, a2, false, be, (short)0, accVar, false, false);
        }
    }

    // B was column-broadcast => every column of D holds the per-row answer.
    // C/D layout: vgpr r, lanes 0-15 -> M=r ; lanes 16-31 -> M=r+8.
    // Lane 0 spills rows 0-7, lane 16 spills rows 8-15.
    if ((lane & 15) == 0) {
        #pragma unroll
        for (int r = 0; r < 8; ++r) {
            meanv[(hi << 3) + r] = accMu[r];
            varv [(hi << 3) + r] = accVar[r];
        }
    }
    __builtin_amdgcn_wave_barrier();   // same-wave LDS RAW; hardware DScnt waits inserted by backend

    // Epilogue: act[m, :] = sqrt(var_m) * z + mean_m ; Y = sigmoid(act).
    const float4 z0 = *(const float4*)(z + lane * 4);          // cols   0..127
    const float4 z1 = *(const float4*)(z + 128 + lane * 4);    // cols 128..255

    #pragma unroll 4
    for (int m = 0; m < 16; ++m) {
        const float  mu   = meanv[m];
        const float  sd   = sqrtf(varv[m]);
        const size_t base = (size_t)(tileRow + m) * NS;

        const float4 a0 = affine4(sd, z0, mu);
        const float4 a1 = affine4(sd, z1, mu);
        const float4 y0 = sig4(a0);
        const float4 y1 = sig4(a1);

        *(float4*)(Act + base + lane * 4)       = a0;
        *(float4*)(Act + base + 128 + lane * 4) = a1;
        *(float4*)(Y   + base + lane * 4)       = y0;
        *(float4*)(Y   + base + 128 + lane * 4) = y1;
    }
}

extern "C" void kernel_launch(void* const* d_in, const int* in_sizes, int n_in,
                              void* d_out, int out_size, void* d_ws, size_t ws_size,
                              hipStream_t stream) {
    const float* X   = (const float*)d_in[0];
    const float* z   = (const float*)d_in[1];
    const float* wmu = (const float*)d_in[2];
    const float* wlv = (const float*)d_in[3];

    float* Yout = (float*)d_out;                      // reference returns (Y, activation_mat)
    float* Act  = Yout + (size_t)N_PTS * NS;
    float* Ew   = (float*)d_ws;                       // 1024 floats of scratch

    expw_kernel<<<(DIM + 255) / 256, 256, 0, stream>>>(wlv, Ew);
    mf_logreg_kernel<<<N_PTS / (WAVES_PER_BLOCK * 16), WAVES_PER_BLOCK * 32, 0, stream>>>(
        X, z, wmu, Ew, Yout, Act);
}